// CapsNet_58875411693957
// MI455X (gfx1250) — compile-verified
//
#include <hip/hip_runtime.h>
#include <hip/hip_bf16.h>

// ---------------------------------------------------------------------------
// CapsNet forward on MI455X (gfx1250), fp32 via V_WMMA_F32_16X16X4_F32.
//
// WMMA f32 16x16x4 fragment layout (CDNA5 ISA 7.12.2):
//   A (16x4):  lanes 0-15: M=lane, V0=K0,V1=K1 ; lanes 16-31: M=lane-16, V0=K2,V1=K3
//   B (4x16):  lanes 0-15: N=lane, V0=K0,V1=K1 ; lanes 16-31: N=lane-16, V0=K2,V1=K3
//   C/D 16x16: VGPR r: lanes 0-15 -> (M=r, N=lane), lanes 16-31 -> (M=r+8, N=lane-16)
//
// K-order for conv2 im2col is (ky, kx, cin): since 256 % 4 == 0, K-steps of 4
// never cross a (ky,kx) boundary -> guard-free, division-free inner loop with
// b64 fragment loads from NHWC activations and (r,n,cin)-packed weights.
// ---------------------------------------------------------------------------

typedef __attribute__((ext_vector_type(2))) float v2f;
typedef __attribute__((ext_vector_type(8))) float v8f;

__device__ __forceinline__ v8f wmma_f32(v2f a, v2f b, v8f c) {
  return __builtin_amdgcn_wmma_f32_16x16x4_f32(false, a, false, b, (short)0, c,
                                               false, false);
}

#define B_SZ 512

// ---------------------------------------------------------------------------
// weight repacks (one-time, trivial vs 196 GFLOP of conv2)
// ---------------------------------------------------------------------------
__global__ __launch_bounds__(256) void repack_w1(const float* __restrict__ w,
                                                 float* __restrict__ w1t) {
  int idx = blockIdx.x * 256 + threadIdx.x;  // (n, r) -> (r, n), 256*81
  if (idx >= 256 * 81) return;
  int n = idx / 81, r = idx % 81;
  w1t[r * 256 + n] = w[idx];
}

__global__ __launch_bounds__(256) void repack_w2(const float* __restrict__ w,
                                                 float* __restrict__ w2t) {
  int idx = blockIdx.x * 256 + threadIdx.x;  // (n, cin, r) -> (r, n, cin)
  if (idx >= 256 * 256 * 81) return;
  int r = idx % 81;
  int t = idx / 81;
  int cin = t & 255;
  int n = t >> 8;
  w2t[((size_t)(r * 256 + n) << 8) + cin] = w[idx];
}

// ---------------------------------------------------------------------------
// conv1: x(512,1,28,28) * w1t(81,256) -> h1t(M=204800, 256) NHWC row-major
// K restructured as ky(0..8) x kx-steps{0,4,8}; invalid kx lanes are
// clamp-addressed + value-selected to 0 (branchless, EXEC stays all-ones).
// ---------------------------------------------------------------------------
__global__ __launch_bounds__(256) void conv1_wmma(
    const float* __restrict__ x, const float* __restrict__ w1t,
    const float* __restrict__ bias, float* __restrict__ h1t) {
  const int tid  = threadIdx.x;
  const int wid  = tid >> 5;
  const int lane = tid & 31;
  const int half = lane >> 4;
  const int l    = lane & 15;

  const int tile = blockIdx.x * 8 + wid;  // 204800 tiles
  const int mt = tile >> 4;
  const int nt = tile & 15;

  const int m  = mt * 16 + l;
  const int bi = m / 400;
  const int p  = m % 400;
  const int oy = p / 20;
  const int ox = p % 20;
  const float* xrow = x + bi * 784 + oy * 28 + ox;

  const int nc = nt * 16 + l;

  v8f acc = {};
  for (int ky = 0; ky < 9; ++ky) {
    const float* xr = xrow + ky * 28;
    const float* wr = w1t + ky * 9 * 256 + nc;
#pragma unroll
    for (int kxs = 0; kxs < 12; kxs += 4) {
      int kx0 = kxs + half * 2;
      int kx1 = kx0 + 1;
      int c0 = kx0 < 9 ? kx0 : 8;   // clamped in-bounds address
      int c1 = kx1 < 9 ? kx1 : 8;
      float a0 = xr[c0];
      float a1 = xr[c1];
      float b0 = wr[c0 * 256];
      float b1 = wr[c1 * 256];
      if (kx0 >= 9) { a0 = 0.f; b0 = 0.f; }   // v_cndmask, no EXEC divergence
      if (kx1 >= 9) { a1 = 0.f; b1 = 0.f; }
      v2f af = {a0, a1};
      v2f bf = {b0, b1};
      acc = wmma_f32(af, bf, acc);
    }
  }

  const float bb = bias[nc];
#pragma unroll
  for (int r = 0; r < 8; ++r) {
    int mr = mt * 16 + r + half * 8;
    h1t[(size_t)mr * 256 + nc] = acc[r] + bb;   // coalesced (nc = consecutive lanes)
  }
}

// ---------------------------------------------------------------------------
// conv2: h1t(204800,256) * w2t(81,256,256) stride2 -> h2(M=18432, 256)
// 2x2 register tiling: one wave owns a 32x32 output tile (4 v8f accumulators);
// A/B fragments shared across the tile row/col -> ~1 b64 load per WMMA.
// Output row-major (m,256) == flattened (B,1152,8) pricaps layout.
// ---------------------------------------------------------------------------
__global__ __launch_bounds__(256) void conv2_wmma(
    const float* __restrict__ h1t, const float* __restrict__ w2t,
    const float* __restrict__ bias, float* __restrict__ h2) {
  const int tid  = threadIdx.x;
  const int wid  = tid >> 5;
  const int lane = tid & 31;
  const int half = lane >> 4;
  const int l    = lane & 15;

  const int job = blockIdx.x * 8 + wid;  // (18432/32)*(256/32) = 4608 jobs
  const int mtp = job >> 3;
  const int ntp = job & 7;

  // two A rows per lane (M-tile pair)
  const int m0 = mtp * 32 + l;
  const int m1 = m0 + 16;
  int bi0 = m0 / 36, p0 = m0 % 36;
  int bi1 = m1 / 36, p1 = m1 % 36;
  const float* a0base =
      h1t + ((size_t)(bi0 * 400 + (p0 / 6) * 40 + (p0 % 6) * 2) << 8);
  const float* a1base =
      h1t + ((size_t)(bi1 * 400 + (p1 / 6) * 40 + (p1 % 6) * 2) << 8);

  const int n0 = ntp * 32 + l;
  const int n1 = n0 + 16;

  v8f acc00 = {}, acc01 = {}, acc10 = {}, acc11 = {};
  for (int ky = 0; ky < 9; ++ky) {
    for (int kx = 0; kx < 9; ++kx) {
      const int r = ky * 9 + kx;
      const float* a0 = a0base + ((ky * 20 + kx) << 8) + half * 2;
      const float* a1 = a1base + ((ky * 20 + kx) << 8) + half * 2;
      const float* b0 = w2t + ((size_t)(r * 256 + n0) << 8) + half * 2;
      const float* b1 = w2t + ((size_t)(r * 256 + n1) << 8) + half * 2;
      __builtin_prefetch(b0 + 256 * 256, 0, 0);  // next r's weight slab
      for (int c = 0; c < 256; c += 4) {
        v2f fa0 = *(const v2f*)(a0 + c);
        v2f fa1 = *(const v2f*)(a1 + c);
        v2f fb0 = *(const v2f*)(b0 + c);
        v2f fb1 = *(const v2f*)(b1 + c);
        acc00 = wmma_f32(fa0, fb0, acc00);
        acc01 = wmma_f32(fa0, fb1, acc01);
        acc10 = wmma_f32(fa1, fb0, acc10);
        acc11 = wmma_f32(fa1, fb1, acc11);
      }
    }
  }

  const float bb0 = bias[n0], bb1 = bias[n1];
#pragma unroll
  for (int r = 0; r < 8; ++r) {
    int mr0 = mtp * 32 + r + half * 8;
    int mr1 = mr0 + 16;
    h2[(size_t)mr0 * 256 + n0] = acc00[r] + bb0;
    h2[(size_t)mr0 * 256 + n1] = acc01[r] + bb1;
    h2[(size_t)mr1 * 256 + n0] = acc10[r] + bb0;
    h2[(size_t)mr1 * 256 + n1] = acc11[r] + bb1;
  }
}

// ---------------------------------------------------------------------------
// squash: h2 rows are already (B,1152,8) flat -> contiguous float4 loads.
// ---------------------------------------------------------------------------
__global__ __launch_bounds__(256) void squash_pricaps(
    const float* __restrict__ h2, float* __restrict__ pri) {
  int idx = blockIdx.x * 256 + threadIdx.x;  // B*1152 capsules
  if (idx >= B_SZ * 1152) return;
  const float4* src = (const float4*)(h2 + (size_t)idx * 8);
  float4 v0 = src[0], v1 = src[1];
  float nsq = v0.x * v0.x + v0.y * v0.y + v0.z * v0.z + v0.w * v0.w +
              v1.x * v1.x + v1.y * v1.y + v1.z * v1.z + v1.w * v1.w;
  float nrm   = sqrtf(nsq);
  float s = nsq / (1.f + nsq) / (nrm + 1e-8f);
  float4* dst = (float4*)(pri + (size_t)idx * 8);
  dst[0] = make_float4(s * v0.x, s * v0.y, s * v0.z, s * v0.w);
  dst[1] = make_float4(s * v1.x, s * v1.y, s * v1.z, s * v1.w);
}

// ---------------------------------------------------------------------------
// Dynamic routing (3 iters) + length + argmax mask.  One block per batch elem.
// hat[o,i,d] recomputed on the fly; route_w (5.9 MB) is L2-resident.
// ---------------------------------------------------------------------------
__global__ __launch_bounds__(256) void routing_kernel(
    const float* __restrict__ pri, const float* __restrict__ rw,
    float* __restrict__ blog, float* __restrict__ cbuf,
    float* __restrict__ lenout, float* __restrict__ ocout) {
  __shared__ float spri[1152 * 8];  // 36 KB
  __shared__ float outc[10][16];
  __shared__ float slen[10];
  __shared__ int   sidx;

  const int b    = blockIdx.x;
  const int tid  = threadIdx.x;
  const int wid  = tid >> 5;
  const int lane = tid & 31;

  float* bl = blog + (size_t)b * 11520;
  float* cc = cbuf + (size_t)b * 11520;

  for (int j = tid; j < 9216; j += 256) spri[j] = pri[(size_t)b * 9216 + j];
  for (int j = tid; j < 11520; j += 256) bl[j] = 0.f;
  __syncthreads();

  for (int it = 0; it < 3; ++it) {
    // c = softmax over o of b_log
    for (int i = tid; i < 1152; i += 256) {
      float v[10];
      float mx = -1e30f;
#pragma unroll
      for (int o = 0; o < 10; ++o) { v[o] = bl[o * 1152 + i]; mx = fmaxf(mx, v[o]); }
      float s = 0.f;
#pragma unroll
      for (int o = 0; o < 10; ++o) { v[o] = __expf(v[o] - mx); s += v[o]; }
      float inv = 1.f / s;
#pragma unroll
      for (int o = 0; o < 10; ++o) cc[o * 1152 + i] = v[o] * inv;
    }
    __syncthreads();

    // s[d] = sum_i c*hat ; squash -> outc
    for (int o = wid; o < 10; o += 8) {
      float s[16];
#pragma unroll
      for (int d = 0; d < 16; ++d) s[d] = 0.f;
      for (int i = lane; i < 1152; i += 32) {
        const float* pv = &spri[i * 8];
        const float* wv = rw + ((size_t)o * 1152 + i) * 128;
        float cv = cc[o * 1152 + i];
#pragma unroll
        for (int d = 0; d < 16; ++d) {
          float h = 0.f;
#pragma unroll
          for (int k = 0; k < 8; ++k) h += wv[d * 8 + k] * pv[k];
          s[d] += cv * h;
        }
      }
#pragma unroll
      for (int d = 0; d < 16; ++d)
        for (int off = 16; off > 0; off >>= 1) s[d] += __shfl_xor(s[d], off, 32);
      if (lane == 0) {
        float nsq = 0.f;
#pragma unroll
        for (int d = 0; d < 16; ++d) nsq += s[d] * s[d];
        float nrm   = sqrtf(nsq);
        float scale = nsq / (1.f + nsq) / (nrm + 1e-8f);
#pragma unroll
        for (int d = 0; d < 16; ++d) outc[o][d] = scale * s[d];
      }
    }
    __syncthreads();

    // b_log += <outc, hat>
    if (it < 2) {
      for (int o = wid; o < 10; o += 8) {
        float ocv[16];
#pragma unroll
        for (int d = 0; d < 16; ++d) ocv[d] = outc[o][d];
        for (int i = lane; i < 1152; i += 32) {
          const float* pv = &spri[i * 8];
          const float* wv = rw + ((size_t)o * 1152 + i) * 128;
          float acc = 0.f;
#pragma unroll
          for (int d = 0; d < 16; ++d) {
            float h = 0.f;
#pragma unroll
            for (int k = 0; k < 8; ++k) h += wv[d * 8 + k] * pv[k];
            acc += ocv[d] * h;
          }
          bl[o * 1152 + i] += acc;
        }
      }
      __syncthreads();
    }
  }

  if (tid < 10) {
    float nsq = 0.f;
#pragma unroll
    for (int d = 0; d < 16; ++d) nsq += outc[tid][d] * outc[tid][d];
    float ln = sqrtf(nsq);
    slen[tid] = ln;
    lenout[b * 10 + tid] = ln;
  }
  __syncthreads();
  if (tid == 0) {
    int best = 0;
    float bv = slen[0];
    for (int o = 1; o < 10; ++o)
      if (slen[o] > bv) { bv = slen[o]; best = o; }
    sidx = best;
  }
  __syncthreads();
  if (tid < 160) {
    int o = tid >> 4, d = tid & 15;
    ocout[b * 160 + tid] = (o == sidx) ? outc[o][d] : 0.f;
  }
}

// ---------------------------------------------------------------------------
// FC: out(M,N) = act(A(M,K) @ W(N,K)^T + bias); aligned v2f fragment loads.
// ---------------------------------------------------------------------------
template <int ACT>
__global__ __launch_bounds__(256) void fc_wmma(
    const float* __restrict__ A, const float* __restrict__ W,
    const float* __restrict__ bias, float* __restrict__ out,
    int M, int N, int K) {
  const int tid  = threadIdx.x;
  const int wid  = tid >> 5;
  const int lane = tid & 31;
  const int half = lane >> 4;
  const int l    = lane & 15;

  const int nTilesN = N >> 4;
  const int tile = blockIdx.x * 8 + wid;
  if (tile >= (M >> 4) * nTilesN) return;  // wave-uniform
  const int mt = tile / nTilesN;
  const int nt = tile % nTilesN;

  const int m  = mt * 16 + l;
  const int nc = nt * 16 + l;
  const float* arow = A + (size_t)m * K + half * 2;
  const float* wrow = W + (size_t)nc * K + half * 2;

  v8f acc = {};
  for (int k0 = 0; k0 < K; k0 += 4) {
    v2f af = *(const v2f*)(arow + k0);
    v2f bf = *(const v2f*)(wrow + k0);
    acc = wmma_f32(af, bf, acc);
  }

  const float bb = bias[nc];
#pragma unroll
  for (int r = 0; r < 8; ++r) {
    int mr = mt * 16 + r + half * 8;
    float vr = acc[r] + bb;
    if (ACT == 0) {
      vr = fmaxf(vr, 0.f);
    } else {
      vr = 1.f / (1.f + __expf(-vr));
      vr = fminf(fmaxf(vr, 0.f), 1.f);
    }
    out[(size_t)mr * N + nc] = vr;
  }
}

// ---------------------------------------------------------------------------
// Launch
// ---------------------------------------------------------------------------
extern "C" void kernel_launch(void* const* d_in, const int* in_sizes, int n_in,
                              void* d_out, int out_size, void* d_ws,
                              size_t ws_size, hipStream_t stream) {
  const float* x   = (const float*)d_in[0];
  const float* c1w = (const float*)d_in[1];
  const float* c1b = (const float*)d_in[2];
  const float* c2w = (const float*)d_in[3];
  const float* c2b = (const float*)d_in[4];
  const float* rw  = (const float*)d_in[5];
  const float* d1w = (const float*)d_in[6];
  const float* d1b = (const float*)d_in[7];
  const float* d2w = (const float*)d_in[8];
  const float* d2b = (const float*)d_in[9];
  const float* d3w = (const float*)d_in[10];
  const float* d3b = (const float*)d_in[11];
  float* out = (float*)d_out;  // [0,5120) objcaps_len, [5120,406528) x_recon

  // workspace layout (floats), ~80M floats (~320 MB)
  float* ws   = (float*)d_ws;
  float* h1t  = ws;                        // 204800*256 = 52,428,800
  float* h2   = h1t + 52428800;            // 18432*256  =  4,718,592
  float* pri  = h2 + 4718592;              // 512*1152*8 =  4,718,592
  float* blog = pri + 4718592;             // 512*10*1152 = 5,898,240
  float* cbuf = blog + 5898240;            // 5,898,240
  float* oc   = cbuf + 5898240;            // 81,920
  float* dd1  = oc + 81920;                // 262,144
  float* dd2  = dd1 + 262144;              // 524,288
  float* w1t  = dd2 + 524288;              // 81*256 = 20,736
  float* w2t  = w1t + 20736;               // 81*256*256 = 5,308,416

  repack_w1<<<(256 * 81 + 255) / 256, 256, 0, stream>>>(c1w, w1t);
  repack_w2<<<(256 * 256 * 81 + 255) / 256, 256, 0, stream>>>(c2w, w2t);

  conv1_wmma<<<25600, 256, 0, stream>>>(x, w1t, c1b, h1t);   // 204800 tiles / 8
  conv2_wmma<<<576, 256, 0, stream>>>(h1t, w2t, c2b, h2);    // 4608 jobs / 8
  squash_pricaps<<<(B_SZ * 1152 + 255) / 256, 256, 0, stream>>>(h2, pri);
  routing_kernel<<<B_SZ, 256, 0, stream>>>(pri, rw, blog, cbuf, out, oc);

  fc_wmma<0><<<128, 256, 0, stream>>>(oc, d1w, d1b, dd1, 512, 512, 160);
  fc_wmma<0><<<256, 256, 0, stream>>>(dd1, d2w, d2b, dd2, 512, 1024, 512);
  fc_wmma<1><<<196, 256, 0, stream>>>(dd2, d3w, d3b, out + 5120, 512, 784, 1024);
}